// PinSageConv_58978490908744
// MI455X (gfx1250) — compile-verified
//
#include <hip/hip_runtime.h>
#include <math.h>

#define N_SRC   100000
#define N_DST   50000
#define N_EDGES 600000
#define EDIM    128

typedef __attribute__((ext_vector_type(2))) float v2f;
typedef __attribute__((ext_vector_type(8))) float v8f;

__device__ __forceinline__ v8f wmma4(v2f a, v2f b, v8f c) {
  // V_WMMA_F32_16X16X4_F32 : D = A(16x4) * B(4x16) + C(16x16), all f32
  return __builtin_amdgcn_wmma_f32_16x16x4_f32(
      /*neg_a=*/false, a, /*neg_b=*/false, b,
      /*c_mod=*/(short)0, c, /*reuse_a=*/false, /*reuse_b=*/false);
}

// ---------------------------------------------------------------------------
// Zero the agg + ws scratch region (must re-zero every launch).
// ---------------------------------------------------------------------------
__global__ void zero_kernel(float* __restrict__ p, size_t n) {
  size_t i = (size_t)blockIdx.x * blockDim.x + threadIdx.x;
  if (i < n) p[i] = 0.0f;
}

// ---------------------------------------------------------------------------
// GEMM1: n = relu(h_src @ q_w.T + q_b)   [N_SRC x 128] = [N_SRC x 128][128 x 128]
// One wave (32 threads) computes a 16-row stripe x all 128 cols.
// ---------------------------------------------------------------------------
__global__ __launch_bounds__(32) void gemm1_kernel(
    const float* __restrict__ h, const float* __restrict__ qw,
    const float* __restrict__ qb, float* __restrict__ nout) {
  const int m0   = blockIdx.x * 16;
  const int lane = threadIdx.x;          // 0..31
  const int l    = lane & 15;
  const int koff = (lane >> 4) * 2;      // K-pair select: 0 or 2

  const float* arow = h  + (size_t)(m0 + l) * EDIM;  // A row m = m0 + l
  const float* brow = qw + (size_t)l * EDIM;         // B col n = nt*16 + l -> q_w row

  v8f acc[8];
#pragma unroll
  for (int i = 0; i < 8; ++i) acc[i] = (v8f){};

  for (int k = 0; k < EDIM; k += 4) {
    v2f a = *(const v2f*)(arow + k + koff);          // h[m][k+koff .. +1]
#pragma unroll
    for (int nt = 0; nt < 8; ++nt) {
      v2f b = *(const v2f*)(brow + (size_t)nt * 16 * EDIM + k + koff); // q_w[n][k..]
      acc[nt] = wmma4(a, b, acc[nt]);
    }
  }

  const int mbase = m0 + ((lane >> 4) * 8);          // C layout M split
#pragma unroll
  for (int nt = 0; nt < 8; ++nt) {
    const int n   = nt * 16 + l;
    const float bias = qb[n];
#pragma unroll
    for (int r = 0; r < 8; ++r) {
      float v = acc[nt][r] + bias;
      v = v > 0.0f ? v : 0.0f;                       // relu
      nout[(size_t)(mbase + r) * EDIM + n] = v;
    }
  }
}

// ---------------------------------------------------------------------------
// Edge scatter: agg[dst] += n[src] * w ; ws[dst] += w   (L2-resident atomics)
// 128 threads per edge, 2 edges per 256-thread block.
// ---------------------------------------------------------------------------
__global__ __launch_bounds__(256) void scatter_kernel(
    const float* __restrict__ nfeat, const float* __restrict__ wts,
    const int* __restrict__ src, const int* __restrict__ dst,
    float* __restrict__ agg, float* __restrict__ wsum) {
  const int e = blockIdx.x * 2 + (threadIdx.x >> 7);
  if (e >= N_EDGES) return;
  const int t = threadIdx.x & 127;
  const float w = wts[e];
  const int s = src[e];
  const int d = dst[e];
  const float v = nfeat[(size_t)s * EDIM + t] * w;
  __hip_atomic_fetch_add(&agg[(size_t)d * EDIM + t], v,
                         __ATOMIC_RELAXED, __HIP_MEMORY_SCOPE_AGENT);
  if (t == 0)
    __hip_atomic_fetch_add(&wsum[d], w,
                           __ATOMIC_RELAXED, __HIP_MEMORY_SCOPE_AGENT);
}

// ---------------------------------------------------------------------------
// GEMM2 + row L2-normalize:
//   z = concat(h_src[:N_DST], agg/clip(ws,1)) : [N_DST x 256]
//   out = normalize_rows(relu(z @ w_w.T + w_b)) : [N_DST x 128]
// ---------------------------------------------------------------------------
__global__ __launch_bounds__(32) void gemm2_kernel(
    const float* __restrict__ h, const float* __restrict__ agg,
    const float* __restrict__ wsum, const float* __restrict__ ww,
    const float* __restrict__ wb, float* __restrict__ out) {
  const int m0   = blockIdx.x * 16;
  const int lane = threadIdx.x;
  const int l    = lane & 15;
  const int koff = (lane >> 4) * 2;
  const int mrow = m0 + l;

  const float* arow1 = h   + (size_t)mrow * EDIM;    // k in [0,128): h_dst
  const float* arow2 = agg + (size_t)mrow * EDIM;    // k in [128,256): agg / ws
  float wsv = wsum[mrow];
  const float inv = 1.0f / (wsv > 1.0f ? wsv : 1.0f);
  const float* brow = ww + (size_t)l * 256;          // w_w row n, stride 256

  v8f acc[8];
#pragma unroll
  for (int i = 0; i < 8; ++i) acc[i] = (v8f){};

  for (int k = 0; k < 128; k += 4) {                 // first half: h_dst
    v2f a = *(const v2f*)(arow1 + k + koff);
#pragma unroll
    for (int nt = 0; nt < 8; ++nt) {
      v2f b = *(const v2f*)(brow + (size_t)nt * 16 * 256 + k + koff);
      acc[nt] = wmma4(a, b, acc[nt]);
    }
  }
  for (int k = 0; k < 128; k += 4) {                 // second half: agg * inv
    v2f a = *(const v2f*)(arow2 + k + koff);
    a.x *= inv; a.y *= inv;
#pragma unroll
    for (int nt = 0; nt < 8; ++nt) {
      v2f b = *(const v2f*)(brow + (size_t)nt * 16 * 256 + 128 + k + koff);
      acc[nt] = wmma4(a, b, acc[nt]);
    }
  }

  // bias + relu (in place), accumulate per-row squared sums
  float ss[8];
#pragma unroll
  for (int r = 0; r < 8; ++r) ss[r] = 0.0f;
#pragma unroll
  for (int nt = 0; nt < 8; ++nt) {
    const int n = nt * 16 + l;
    const float bias = wb[n];
#pragma unroll
    for (int r = 0; r < 8; ++r) {
      float v = acc[nt][r] + bias;
      v = v > 0.0f ? v : 0.0f;
      acc[nt][r] = v;
      ss[r] += v * v;
    }
  }

  // reduce across the 16 lanes of each half-wave (masks 1,2,4,8 stay in-half)
#pragma unroll
  for (int mask = 1; mask < 16; mask <<= 1) {
#pragma unroll
    for (int r = 0; r < 8; ++r) ss[r] += __shfl_xor(ss[r], mask, 32);
  }

  float inorm[8];
#pragma unroll
  for (int r = 0; r < 8; ++r) {
    float nn = sqrtf(ss[r]);
    inorm[r] = nn > 0.0f ? 1.0f / nn : 1.0f;
  }

  const int mbase = m0 + ((lane >> 4) * 8);
#pragma unroll
  for (int nt = 0; nt < 8; ++nt) {
    const int n = nt * 16 + l;
#pragma unroll
    for (int r = 0; r < 8; ++r)
      out[(size_t)(mbase + r) * EDIM + n] = acc[nt][r] * inorm[r];
  }
}

// ---------------------------------------------------------------------------
extern "C" void kernel_launch(void* const* d_in, const int* in_sizes, int n_in,
                              void* d_out, int out_size, void* d_ws, size_t ws_size,
                              hipStream_t stream) {
  const float* h_src   = (const float*)d_in[0];
  const float* weights = (const float*)d_in[1];
  const int*   src_idx = (const int*)d_in[2];
  const int*   dst_idx = (const int*)d_in[3];
  // d_in[4] = num_dst (fixed at N_DST)
  const float* q_w = (const float*)d_in[5];
  const float* q_b = (const float*)d_in[6];
  const float* w_w = (const float*)d_in[7];
  const float* w_b = (const float*)d_in[8];
  float* out = (float*)d_out;

  // workspace layout: [ nfeat (N_SRC*128) | agg (N_DST*128) | ws (N_DST) ]
  float* nfeat = (float*)d_ws;
  float* agg   = nfeat + (size_t)N_SRC * EDIM;
  float* wsum  = agg   + (size_t)N_DST * EDIM;

  const size_t zn = (size_t)N_DST * EDIM + N_DST;
  zero_kernel<<<(unsigned)((zn + 255) / 256), 256, 0, stream>>>(agg, zn);

  gemm1_kernel<<<N_SRC / 16, 32, 0, stream>>>(h_src, q_w, q_b, nfeat);

  scatter_kernel<<<(N_EDGES + 1) / 2, 256, 0, stream>>>(
      nfeat, weights, src_idx, dst_idx, agg, wsum);

  gemm2_kernel<<<N_DST / 16, 32, 0, stream>>>(h_src, agg, wsum, w_w, w_b, out);
}